// AlignedLinear_73469710566097
// MI455X (gfx1250) — compile-verified
//
#include <hip/hip_runtime.h>

// CDNA5 / gfx1250: fp32 WMMA path for the irrep-aligned linear layer.
// y[n, o*9+m] = alpha * sum_i x[n, i*9+m] * K[irrep(m), i, o]
// irreps: m=0 -> K[0], m=1..3 -> K[1], m=4..8 -> K[2]

typedef __attribute__((ext_vector_type(2))) float v2f;
typedef __attribute__((ext_vector_type(8))) float v8f;

#define DIMT 9                 // total representation dim (1+3+5)
#define MUL  128               // mul_in == mul_out
#define ROW  (MUL * DIMT)      // 1152 floats per node row
#define NODES_PER_BLOCK 16
#define THREADS 256            // 8 wave32
#define KTILES 32              // K=128 in steps of 4 (wmma f32 16x16x4)
#define N_NODES 50000

__global__ void __launch_bounds__(THREADS)
aligned_linear_wmma(const float* __restrict__ x,
                    const float* __restrict__ kern,
                    float* __restrict__ y,
                    float alpha)
{
    extern __shared__ float lds[];
    float* xs = lds;                                // 16*1152 floats (72 KB)
    float* ys = lds + NODES_PER_BLOCK * ROW;        // 16*1152 floats (72 KB)

    const int tid = threadIdx.x;
    const long long n0 = (long long)blockIdx.x * NODES_PER_BLOCK;

    // ---- stage X tile: fully coalesced 128-bit copies ----
    {
        const float4* src = (const float4*)(x + n0 * ROW);
        float4* dst = (float4*)xs;
        const int nvec = NODES_PER_BLOCK * ROW / 4;   // 4608
        for (int i = tid; i < nvec; i += THREADS) dst[i] = src[i];
    }
    __syncthreads();

    const int wave = tid >> 5;       // 0..7 -> output-channel tile
    const int lane = tid & 31;
    const int half = lane >> 4;      // 0/1 (K sub-pair per ISA layout)
    const int mrow = lane & 15;      // A: M row; B: N column
    const int o0   = wave * 16;

    const int mstart[4] = {0, 1, 4, 9};   // irrep boundaries in m

    for (int g = 0; g < 3; ++g) {
        // Preload full-K B slab for this irrep & o-tile into registers.
        // B(4x16) layout: VGPR v, lanes[0:16) -> K = 4*kt + v + 2*half, N = mrow
        v2f b[KTILES];
        const float* kg = kern + (size_t)g * MUL * MUL + o0 + mrow;
#pragma unroll
        for (int kt = 0; kt < KTILES; ++kt) {
            const int kb = kt * 4 + half * 2;
            b[kt].x = kg[(kb + 0) * MUL];
            b[kt].y = kg[(kb + 1) * MUL];
        }

        for (int m = mstart[g]; m < mstart[g + 1]; ++m) {
            v8f c = {};
            // A(16x4) layout: lane%16 = M, VGPR v -> K = 4*kt + v + 2*half
            const float* xa = xs + mrow * ROW + m;
#pragma unroll
            for (int kt = 0; kt < KTILES; ++kt) {
                const int kb = kt * 4 + half * 2;
                v2f a;
                a.x = xa[(kb + 0) * DIMT];
                a.y = xa[(kb + 1) * DIMT];
                c = __builtin_amdgcn_wmma_f32_16x16x4_f32(
                        /*neg_a=*/false, a, /*neg_b=*/false, b[kt],
                        /*c_mod=*/(short)0, c,
                        /*reuse_a=*/false, /*reuse_b=*/false);
            }
            // C(16x16) layout: VGPR r -> M = r + 8*half, N = mrow
            float* yb = ys + (o0 + mrow) * DIMT + m;
#pragma unroll
            for (int r = 0; r < 8; ++r) {
                const int node = r + half * 8;
                yb[node * ROW] = alpha * c[r];
            }
        }
    }
    __syncthreads();

    // ---- writeback Y tile: fully coalesced 128-bit stores ----
    {
        float4* dst = (float4*)(y + n0 * ROW);
        const float4* src = (const float4*)ys;
        const int nvec = NODES_PER_BLOCK * ROW / 4;
        for (int i = tid; i < nvec; i += THREADS) dst[i] = src[i];
    }
}

extern "C" void kernel_launch(void* const* d_in, const int* in_sizes, int n_in,
                              void* d_out, int out_size, void* d_ws, size_t ws_size,
                              hipStream_t stream) {
    const float* x    = (const float*)d_in[0];   // [50000, 1152] fp32
    const float* kern = (const float*)d_in[1];   // [3, 128, 128] fp32
    float* y          = (float*)d_out;           // [50000, 1152] fp32

    const float alpha = 0.08838834764831845f;    // 1/sqrt(128)
    const int blocks = N_NODES / NODES_PER_BLOCK;            // 3125 exact
    const size_t lds_bytes = 2u * NODES_PER_BLOCK * ROW * 4; // 144 KB

    aligned_linear_wmma<<<blocks, THREADS, lds_bytes, stream>>>(x, kern, y, alpha);
}